// SpectralVSAArchive_9234179686597
// MI455X (gfx1250) — compile-verified
//
#include <hip/hip_runtime.h>
#include <math.h>
#include <stdint.h>

typedef __attribute__((ext_vector_type(2))) float v2f;
typedef __attribute__((ext_vector_type(8))) float v8f;
typedef int v4i __attribute__((vector_size(16)));
typedef __attribute__((address_space(1))) v4i* gv4i_p;   // global (AS1) 16B vector ptr
typedef __attribute__((address_space(3))) v4i* lv4i_p;   // LDS (AS3) 16B vector ptr

#define DM      2048          // D_MODEL
#define KB      64            // K bands
#define WW      512           // window
#define STR     256           // stride
#define NWIN    32            // S / STRIDE
#define SS      8192          // S
#define BSTRIDE (SS * DM)     // batch stride in scan_out
#define EMA_A   0.9f
#define EMA_B   0.1f
#define EPSF    1e-12f
#define KSTEP   32            // K rows staged per LDS stage

#if __has_builtin(__builtin_amdgcn_global_load_async_to_lds_b128) && \
    __has_builtin(__builtin_amdgcn_s_wait_asynccnt)
#define USE_ASYNC 1
#else
#define USE_ASYNC 0
#endif

// ---------------- workspace layout (floats) ----------------
#define WS_COEFFS 0                         // 32*64*2048 = 4194304
#define WS_NORMS  4194304                   // 32*64 = 2048
#define WS_CNORMS (WS_NORMS + 2048)         // 64
#define WS_P      (WS_CNORMS + 64)          // 64*2048 = 131072
#define WS_CTX    (WS_P + 131072)           // 2048
#define WS_TOP    (WS_CTX + 2048)           // 8 ints (reinterpret)
#define WS_WGT    (WS_TOP + 8)              // 8 floats

#if USE_ASYNC
__device__ __forceinline__ void async_cp16(const float* gp, float* lp) {
    // low 32 bits of a generic LDS pointer are the LDS byte offset
    __builtin_amdgcn_global_load_async_to_lds_b128(
        (gv4i_p)(uintptr_t)gp,
        (lv4i_p)(uint32_t)(uintptr_t)lp,
        0, 0);
}
#endif

// ============================================================
// Kernel A: coeffs[win][m][n] = (1/W) * sum over 1024 K-rows of
//   cheby[m][k%512] * src_b[k], b = k/512 selects batch copy
// (batch-mean folded as 0.5 into output scale; window-0 old half
//  reads buf in both passes -> 0.5+0.5 = 1.0, correct).
// fp32 WMMA 16x16x4. Block = 128 threads (4 waves): wave w owns
// M rows [16w,16w+16), all 64 N columns. B tile async-staged in LDS.
// ============================================================
__global__ __launch_bounds__(128) void kA_cheby_gemm(
    const float* __restrict__ scan_out, const float* __restrict__ buf,
    const float* __restrict__ cheby, float* __restrict__ coeffs)
{
    __shared__ __align__(16) float bs[KSTEP * 64];   // 8 KB
    const int n0   = blockIdx.x * 64;     // N tile
    const int win  = blockIdx.y;          // window
    const int tid  = threadIdx.x;
    const int wave = tid >> 5;
    const int lane = tid & 31;
    const int lm   = lane & 15;
    const int hi   = lane >> 4;
    const int m0   = wave * 16;

    v8f acc0 = {}, acc1 = {}, acc2 = {}, acc3 = {};

    for (int b = 0; b < 2; ++b) {
        const float* srcScan = scan_out + (size_t)b * BSTRIDE;
        for (int k0 = 0; k0 < WW; k0 += KSTEP) {
            // block-uniform source selection:
            //  win==0 && k0<256 -> buf rows 256+k0.. ; else h-rows (win-1)*256+k0..
            const bool useBuf = (win == 0) && (k0 < STR);
            const float* base = useBuf
                ? (buf + (size_t)(STR + k0) * DM)
                : (srcScan + (size_t)((win - 1) * STR + k0) * DM);

            // ---- stage KSTEP x 64 tile: 512 x 16B chunks, 4 per thread ----
            #pragma unroll
            for (int j = 0; j < 4; ++j) {
                const int c    = tid + 128 * j;     // 0..511
                const int row  = c >> 4;            // 0..31
                const int col4 = (c & 15) << 2;     // 0,4,...,60
                const float* gp = base + (size_t)row * DM + n0 + col4;
                float* lp = &bs[row * 64 + col4];
#if USE_ASYNC
                async_cp16(gp, lp);
#else
                *(float4*)lp = *(const float4*)gp;
#endif
            }
#if USE_ASYNC
            __builtin_amdgcn_s_wait_asynccnt(0);
#endif
            __syncthreads();

            // ---- WMMA over the staged KSTEP rows ----
            #pragma unroll
            for (int ks = 0; ks < KSTEP; ks += 4) {
                // A fragment (16x4): lanes 0-15 K=ks..ks+1, lanes 16-31 K=ks+2..ks+3
                const float* arow = cheby + (size_t)(m0 + lm) * WW + (k0 + ks + 2 * hi);
                v2f a; a.x = arow[0]; a.y = arow[1];
                const int r0 = (ks + 2 * hi) * 64;
                const int r1 = r0 + 64;
                v2f b0, b1, b2, b3;
                b0.x = bs[r0 + lm +  0]; b0.y = bs[r1 + lm +  0];
                b1.x = bs[r0 + lm + 16]; b1.y = bs[r1 + lm + 16];
                b2.x = bs[r0 + lm + 32]; b2.y = bs[r1 + lm + 32];
                b3.x = bs[r0 + lm + 48]; b3.y = bs[r1 + lm + 48];
                acc0 = __builtin_amdgcn_wmma_f32_16x16x4_f32(false, a, false, b0, (short)0, acc0, false, false);
                acc1 = __builtin_amdgcn_wmma_f32_16x16x4_f32(false, a, false, b1, (short)0, acc1, false, false);
                acc2 = __builtin_amdgcn_wmma_f32_16x16x4_f32(false, a, false, b2, (short)0, acc2, false, false);
                acc3 = __builtin_amdgcn_wmma_f32_16x16x4_f32(false, a, false, b3, (short)0, acc3, false, false);
            }
            __syncthreads();
        }
    }

    // ---- store: C/D layout lane(l): VGPR r -> (M = r + 8*hi, N = lm) ----
    // scale = 0.5 (batch mean) * 2/W, and row 0 additionally halved
    const float base = 0.5f * 2.0f / (float)WW;
    #pragma unroll
    for (int r = 0; r < 8; ++r) {
        const int m = m0 + r + 8 * hi;
        const float scale = base * ((m == 0) ? 0.5f : 1.0f);
        float* dst = coeffs + ((size_t)win * KB + m) * DM + n0 + lm;
        dst[ 0] = acc0[r] * scale;
        dst[16] = acc1[r] * scale;
        dst[32] = acc2[r] * scale;
        dst[48] = acc3[r] * scale;
    }
}

// ---------------- block reduction helper (256 threads) ----------------
__device__ __forceinline__ float blockReduce256(float s, float* red) {
    #pragma unroll
    for (int off = 16; off > 0; off >>= 1) s += __shfl_xor(s, off, 32);
    if ((threadIdx.x & 31) == 0) red[threadIdx.x >> 5] = s;
    __syncthreads();
    float t = 0.0f;
    if (threadIdx.x == 0) {
        #pragma unroll
        for (int w = 0; w < 8; ++w) t += red[w];
    }
    return t;
}

// norms[i][k] = max(||coeffs[i,k,:]||, EPS)
__global__ __launch_bounds__(256) void kB_norms(const float* __restrict__ coeffs,
                                                float* __restrict__ norms)
{
    __shared__ float red[8];
    const int i = blockIdx.x >> 6;
    const int k = blockIdx.x & 63;
    const float* row = coeffs + ((size_t)i * KB + k) * DM;
    float s = 0.0f;
    for (int d = threadIdx.x; d < DM; d += 256) { float v = row[d]; s += v * v; }
    float t = blockReduce256(s, red);
    if (threadIdx.x == 0) norms[i * KB + k] = fmaxf(sqrtf(t), EPSF);
}

// delta[k] = ||coeffs[31,k,:] - coeffs[30,k,:]||  (last-step delta only)
__global__ __launch_bounds__(256) void kB_delta(const float* __restrict__ coeffs,
                                                float* __restrict__ delta_out)
{
    __shared__ float red[8];
    const int k = blockIdx.x;
    const float* a = coeffs + ((size_t)31 * KB + k) * DM;
    const float* b = coeffs + ((size_t)30 * KB + k) * DM;
    float s = 0.0f;
    for (int d = threadIdx.x; d < DM; d += 256) { float v = a[d] - b[d]; s += v * v; }
    float t = blockReduce256(s, red);
    if (threadIdx.x == 0) delta_out[k] = sqrtf(t);
}

// cnorms[k] = exact EMA replay of norms over 32 windows
__global__ void kC_cnorms(const float* __restrict__ norms, float* __restrict__ cnorms)
{
    const int k = threadIdx.x;
    if (k < KB) {
        float c = 0.0f;
        for (int i = 0; i < NWIN; ++i) c = EMA_A * c + EMA_B * norms[i * KB + k];
        cnorms[k] = c;
    }
}

// P[k][d] = EMA_i( coeffs[i,k,d] / norms[i,k] )
__global__ __launch_bounds__(256) void kC1_P(const float* __restrict__ coeffs,
                                             const float* __restrict__ norms,
                                             float* __restrict__ P)
{
    const int k = blockIdx.x;
    const int d = blockIdx.y * 256 + threadIdx.x;
    float p = 0.0f;
    for (int i = 0; i < NWIN; ++i) {
        const float inv = 1.0f / norms[i * KB + k];
        p = EMA_A * p + EMA_B * coeffs[((size_t)i * KB + k) * DM + d] * inv;
    }
    P[(size_t)k * DM + d] = p;
}

// vr[d] = sum_k P[k,d]*rr[k,d] ; vi likewise
__global__ __launch_bounds__(256) void kC2_vrvi(const float* __restrict__ P,
                                                const float* __restrict__ rr,
                                                const float* __restrict__ ri,
                                                float* __restrict__ vr,
                                                float* __restrict__ vi)
{
    const int d = blockIdx.x * 256 + threadIdx.x;
    float sr = 0.0f, si = 0.0f;
    for (int k = 0; k < KB; ++k) {
        const float p = P[(size_t)k * DM + d];
        sr += p * rr[(size_t)k * DM + d];
        si += p * ri[(size_t)k * DM + d];
    }
    vr[d] = sr;
    vi[d] = si;
}

// top-8 by delta (ties -> lowest index, strictly-greater) + weights
__global__ void kD1_top(const float* __restrict__ delta,
                        const float* __restrict__ cnorms,
                        const float* __restrict__ w_bands,
                        int* __restrict__ topidx, float* __restrict__ wgt)
{
    if (threadIdx.x == 0) {
        bool used[KB];
        for (int k = 0; k < KB; ++k) used[k] = false;
        for (int j = 0; j < 8; ++j) {
            int best = 0; float bv = -INFINITY;
            for (int k = 0; k < KB; ++k)
                if (!used[k] && delta[k] > bv) { bv = delta[k]; best = k; }
            used[best] = true;
            topidx[j] = best;
            const float sp = log1pf(expf(w_bands[best]));   // softplus
            wgt[j] = sp * fmaxf(cnorms[best], EPSF);
        }
    }
}

// ctx[d] = sum_j wgt[j] * (vr[d]*rr[tj,d] + vi[d]*ri[tj,d])
__global__ __launch_bounds__(256) void kD2_ctx(const float* __restrict__ vr,
                                               const float* __restrict__ vi,
                                               const float* __restrict__ rr,
                                               const float* __restrict__ ri,
                                               const int* __restrict__ topidx,
                                               const float* __restrict__ wgt,
                                               float* __restrict__ ctx)
{
    const int d = blockIdx.x * 256 + threadIdx.x;
    const float vrd = vr[d], vid = vi[d];
    float s = 0.0f;
    #pragma unroll
    for (int j = 0; j < 8; ++j) {
        const int k = topidx[j];
        s += wgt[j] * (vrd * rr[(size_t)k * DM + d] + vid * ri[(size_t)k * DM + d]);
    }
    ctx[d] = s;
}

// out[r] = sigmoid(gate) * sum_d ctx[d]*W[r,d]
__global__ __launch_bounds__(256) void kD3_gemv(const float* __restrict__ ctx,
                                                const float* __restrict__ Wm,
                                                const float* __restrict__ gate,
                                                float* __restrict__ out)
{
    __shared__ float red[8];
    const int r = blockIdx.x;
    const float* row = Wm + (size_t)r * DM;
    float s = 0.0f;
    for (int d = threadIdx.x; d < DM; d += 256) s += ctx[d] * row[d];
    float t = blockReduce256(s, red);
    if (threadIdx.x == 0) {
        const float g = 1.0f / (1.0f + expf(-gate[0]));
        out[r] = t * g;
    }
}

// ============================================================
extern "C" void kernel_launch(void* const* d_in, const int* in_sizes, int n_in,
                              void* d_out, int out_size, void* d_ws, size_t ws_size,
                              hipStream_t stream)
{
    const float* scan_out = (const float*)d_in[0];   // [2,8192,2048]
    const float* buf      = (const float*)d_in[1];   // [512,2048]
    const float* rr       = (const float*)d_in[2];   // [64,2048]
    const float* ri       = (const float*)d_in[3];   // [64,2048]
    const float* w_bands  = (const float*)d_in[4];   // [64]
    const float* proj     = (const float*)d_in[5];   // [2048,2048]
    const float* gate     = (const float*)d_in[6];   // [1]
    const float* cheby    = (const float*)d_in[7];   // [64,512]

    float* outf = (float*)d_out;
    float* out_vec   = outf;                // [2048]
    float* out_delta = outf + DM;           // [64]
    float* out_vr    = outf + DM + KB;      // [2048]
    float* out_vi    = outf + DM + KB + DM; // [2048]

    float* ws     = (float*)d_ws;
    float* coeffs = ws + WS_COEFFS;
    float* norms  = ws + WS_NORMS;
    float* cnorms = ws + WS_CNORMS;
    float* P      = ws + WS_P;
    float* ctx    = ws + WS_CTX;
    int*   topidx = (int*)(ws + WS_TOP);
    float* wgt    = ws + WS_WGT;

    // 1) Chebyshev GEMM for all 32 windows (fp32 WMMA, async LDS staging)
    kA_cheby_gemm<<<dim3(DM / 64, NWIN), 128, 0, stream>>>(scan_out, buf, cheby, coeffs);

    // 2) per-window band norms; final delta
    kB_norms<<<NWIN * KB, 256, 0, stream>>>(coeffs, norms);
    kB_delta<<<KB, 256, 0, stream>>>(coeffs, out_delta);

    // 3) EMA replays (linear, exact) -> cnorms, P, vr/vi
    kC_cnorms<<<1, KB, 0, stream>>>(norms, cnorms);
    kC1_P<<<dim3(KB, DM / 256), 256, 0, stream>>>(coeffs, norms, P);
    kC2_vrvi<<<DM / 256, 256, 0, stream>>>(P, rr, ri, out_vr, out_vi);

    // 4) retrieve: top-8, ctx, projection GEMV
    kD1_top<<<1, 32, 0, stream>>>(out_delta, cnorms, w_bands, topidx, wgt);
    kD2_ctx<<<DM / 256, 256, 0, stream>>>(out_vr, out_vi, rr, ri, topidx, wgt, ctx);
    kD3_gemv<<<DM, 256, 0, stream>>>(ctx, proj, gate, out_vec);
}